// MDTA_74045236183622
// MI455X (gfx1250) — compile-verified
//
#include <hip/hip_runtime.h>
#include <hip/hip_bf16.h>
#include <math.h>

typedef float v2f __attribute__((ext_vector_type(2)));
typedef float v8f __attribute__((ext_vector_type(8)));

#define NN 16384      // H*W = 128*128
#define CC 192
#define C3 576
#define BB 8
#define KSPLIT 32

// ---------------------------------------------------------------------------
// GEMM: C[b] = A[b](MxK) * B[b](KxN), K == 192, fp32 WMMA 16x16x4.
// Block: 128 threads (4 waves). Block tile: 32M x 256N. Wave tile: 32M x 64N.
// Two 16-row A tiles staged in LDS (32 x 192 = 24 KB); every B fragment is
// reused by 2 WMMAs (A0-, A1-row sets) -> half the B-side traffic.
// ---------------------------------------------------------------------------
__global__ __launch_bounds__(128)
void gemm_wmma_f32(const float* __restrict__ A, long long aBatchStride,
                   const float* __restrict__ B, long long bBatchStride,
                   float* __restrict__ Cm, long long cBatchStride,
                   int M, int N, int K)
{
    __shared__ float aTile[32 * CC];
    const int b = blockIdx.z;
    const float* Ab = A + (long long)b * aBatchStride;
    const float* Bb = B + (long long)b * bBatchStride;
    float* Cb = Cm + (long long)b * cBatchStride;
    const int m0 = blockIdx.y * 32;

    for (int i = threadIdx.x; i < 32 * K; i += blockDim.x) {
        int r = i / K, c = i - r * K;
        aTile[i] = Ab[(long long)(m0 + r) * K + c];
    }
    __syncthreads();

    const int wave  = threadIdx.x >> 5;
    const int lane  = threadIdx.x & 31;
    const int half  = lane >> 4;     // 0: lanes 0-15, 1: lanes 16-31
    const int lan16 = lane & 15;
    const int klo   = half * 2;      // upper half-lanes carry K+2/K+3
    const int nb    = blockIdx.x * 256 + wave * 64 + lan16;

    v8f acc00 = {}, acc01 = {}, acc02 = {}, acc03 = {};
    v8f acc10 = {}, acc11 = {}, acc12 = {}, acc13 = {};

    for (int kk = 0; kk < K; kk += 4) {
        // A fragments: rows m0+lan16 and m0+16+lan16, K = kk+klo .. +1
        v2f a0 = *(const v2f*)(&aTile[lan16 * K + kk + klo]);
        v2f a1 = *(const v2f*)(&aTile[(16 + lan16) * K + kk + klo]);
        const float* brow0 = Bb + (long long)(kk + klo) * N;
        const float* brow1 = brow0 + N;
        v2f b0; b0.x = brow0[nb];      b0.y = brow1[nb];
        v2f b1; b1.x = brow0[nb + 16]; b1.y = brow1[nb + 16];
        v2f b2; b2.x = brow0[nb + 32]; b2.y = brow1[nb + 32];
        v2f b3; b3.x = brow0[nb + 48]; b3.y = brow1[nb + 48];
        acc00 = __builtin_amdgcn_wmma_f32_16x16x4_f32(false, a0, false, b0, (short)0, acc00, false, false);
        acc01 = __builtin_amdgcn_wmma_f32_16x16x4_f32(false, a0, false, b1, (short)0, acc01, false, false);
        acc02 = __builtin_amdgcn_wmma_f32_16x16x4_f32(false, a0, false, b2, (short)0, acc02, false, false);
        acc03 = __builtin_amdgcn_wmma_f32_16x16x4_f32(false, a0, false, b3, (short)0, acc03, false, false);
        acc10 = __builtin_amdgcn_wmma_f32_16x16x4_f32(false, a1, false, b0, (short)0, acc10, false, false);
        acc11 = __builtin_amdgcn_wmma_f32_16x16x4_f32(false, a1, false, b1, (short)0, acc11, false, false);
        acc12 = __builtin_amdgcn_wmma_f32_16x16x4_f32(false, a1, false, b2, (short)0, acc12, false, false);
        acc13 = __builtin_amdgcn_wmma_f32_16x16x4_f32(false, a1, false, b3, (short)0, acc13, false, false);
    }

    // C/D layout: VGPR r -> row base + r + half*8, col nb (+16 per sub-tile)
    for (int r = 0; r < 8; ++r) {
        float* crow0 = Cb + (long long)(m0 + r + half * 8) * N;
        float* crow1 = Cb + (long long)(m0 + 16 + r + half * 8) * N;
        crow0[nb]      = acc00[r];
        crow0[nb + 16] = acc01[r];
        crow0[nb + 32] = acc02[r];
        crow0[nb + 48] = acc03[r];
        crow1[nb]      = acc10[r];
        crow1[nb + 16] = acc11[r];
        crow1[nb + 32] = acc12[r];
        crow1[nb + 48] = acc13[r];
    }
}

// ---------------------------------------------------------------------------
// Depthwise 3x3, SAME zero padding, one thread per output pixel.
// grid = (NN/256, 576, B)
// ---------------------------------------------------------------------------
__global__ __launch_bounds__(256)
void dwconv3x3(const float* __restrict__ in, const float* __restrict__ wdw,
               float* __restrict__ out)
{
    const int ch = blockIdx.y, b = blockIdx.z;
    const int pix = blockIdx.x * 256 + threadIdx.x;
    const int x = pix & 127, y = pix >> 7;
    const float* plane = in + ((long long)b * C3 + ch) * NN;
    const float* w = wdw + ch * 9;
    float acc = 0.0f;
#pragma unroll
    for (int dy = -1; dy <= 1; ++dy)
#pragma unroll
        for (int dx = -1; dx <= 1; ++dx) {
            int yy = y + dy, xx = x + dx;
            float v = (yy >= 0 && yy < 128 && xx >= 0 && xx < 128)
                          ? plane[yy * 128 + xx] : 0.0f;
            acc = fmaf(v, w[(dy + 1) * 3 + (dx + 1)], acc);
        }
    out[((long long)b * C3 + ch) * NN + pix] = acc;
}

// ---------------------------------------------------------------------------
// Row sum-of-squares for q,k channels (first 384 of qkv_dw). Deterministic
// block reduction, one block per (channel, batch). grid = (384, B)
// ---------------------------------------------------------------------------
__global__ __launch_bounds__(256)
void rowsumsq(const float* __restrict__ qkvdw, float* __restrict__ sumsq)
{
    const int ch = blockIdx.x, b = blockIdx.y;
    const float* row = qkvdw + ((long long)b * C3 + ch) * NN;
    float s = 0.0f;
    for (int i = threadIdx.x; i < NN; i += 256) { float v = row[i]; s = fmaf(v, v, s); }
    __shared__ float red[256];
    red[threadIdx.x] = s; __syncthreads();
    for (int t = 128; t > 0; t >>= 1) {
        if (threadIdx.x < t) red[threadIdx.x] += red[threadIdx.x + t];
        __syncthreads();
    }
    if (threadIdx.x == 0) sumsq[b * 384 + ch] = red[0];
}

// ---------------------------------------------------------------------------
// Gram partials: Spart[ks][b,h][48][48] = q_chunk * k_chunk^T via WMMA f32.
// One wave computes the FULL 48x48 (9 accumulators): 3 A + 3 B fragments per
// K-step feed 9 WMMAs (3x the load:WMMA ratio of a single-tile wave), and
// each q/k element is loaded once. grid = (KSPLIT, B*4), 1 wave per block,
// plain stores into a private split-K slice (no atomics -> deterministic).
// ---------------------------------------------------------------------------
__global__ __launch_bounds__(32)
void gram_wmma(const float* __restrict__ qkvdw, float* __restrict__ Spart)
{
    const int bh = blockIdx.y;
    const int b = bh >> 2, h = bh & 3;
    const int kcLen = NN / KSPLIT;
    const int k0 = blockIdx.x * kcLen;

    const float* Q  = qkvdw + ((long long)b * C3 + h * 48) * NN;
    const float* Km = qkvdw + ((long long)b * C3 + CC + h * 48) * NN;

    const int lane = threadIdx.x & 31;
    const int half = lane >> 4;
    const int lan16 = lane & 15;
    const int klo = half * 2;

    const float* q0 = Q  + (long long)(lan16)      * NN;
    const float* q1 = Q  + (long long)(16 + lan16) * NN;
    const float* q2 = Q  + (long long)(32 + lan16) * NN;
    const float* k0r = Km + (long long)(lan16)      * NN;
    const float* k1r = Km + (long long)(16 + lan16) * NN;
    const float* k2r = Km + (long long)(32 + lan16) * NN;

    v8f c00 = {}, c01 = {}, c02 = {};
    v8f c10 = {}, c11 = {}, c12 = {};
    v8f c20 = {}, c21 = {}, c22 = {};

    for (int kk = k0; kk < k0 + kcLen; kk += 4) {
        v2f a0 = *(const v2f*)(q0 + kk + klo);
        v2f a1 = *(const v2f*)(q1 + kk + klo);
        v2f a2 = *(const v2f*)(q2 + kk + klo);
        v2f b0 = *(const v2f*)(k0r + kk + klo);
        v2f b1 = *(const v2f*)(k1r + kk + klo);
        v2f b2 = *(const v2f*)(k2r + kk + klo);
        c00 = __builtin_amdgcn_wmma_f32_16x16x4_f32(false, a0, false, b0, (short)0, c00, false, false);
        c01 = __builtin_amdgcn_wmma_f32_16x16x4_f32(false, a0, false, b1, (short)0, c01, false, false);
        c02 = __builtin_amdgcn_wmma_f32_16x16x4_f32(false, a0, false, b2, (short)0, c02, false, false);
        c10 = __builtin_amdgcn_wmma_f32_16x16x4_f32(false, a1, false, b0, (short)0, c10, false, false);
        c11 = __builtin_amdgcn_wmma_f32_16x16x4_f32(false, a1, false, b1, (short)0, c11, false, false);
        c12 = __builtin_amdgcn_wmma_f32_16x16x4_f32(false, a1, false, b2, (short)0, c12, false, false);
        c20 = __builtin_amdgcn_wmma_f32_16x16x4_f32(false, a2, false, b0, (short)0, c20, false, false);
        c21 = __builtin_amdgcn_wmma_f32_16x16x4_f32(false, a2, false, b1, (short)0, c21, false, false);
        c22 = __builtin_amdgcn_wmma_f32_16x16x4_f32(false, a2, false, b2, (short)0, c22, false, false);
    }

    float* Sp = Spart + ((long long)blockIdx.x * (BB * 4) + bh) * 48 * 48;
    for (int r = 0; r < 8; ++r) {
        int dr = r + half * 8;
        int e  = lan16;
        Sp[(dr)      * 48 + e]      = c00[r];
        Sp[(dr)      * 48 + e + 16] = c01[r];
        Sp[(dr)      * 48 + e + 32] = c02[r];
        Sp[(dr + 16) * 48 + e]      = c10[r];
        Sp[(dr + 16) * 48 + e + 16] = c11[r];
        Sp[(dr + 16) * 48 + e + 32] = c12[r];
        Sp[(dr + 32) * 48 + e]      = c20[r];
        Sp[(dr + 32) * 48 + e + 16] = c21[r];
        Sp[(dr + 32) * 48 + e + 32] = c22[r];
    }
}

// ---------------------------------------------------------------------------
// Reduce split-K partials, apply L2-norm scaling + temperature, softmax over
// e, then fold output projection: Mcomb[b][o][h*48+e] = sum_d P[o,h*48+d]*A[d,e]
// grid = (B*4), 256 threads.
// ---------------------------------------------------------------------------
__global__ __launch_bounds__(256)
void attn_fold(const float* __restrict__ Spart, const float* __restrict__ sumsq,
               const float* __restrict__ temperature, const float* __restrict__ w_proj,
               float* __restrict__ Mcomb)
{
    __shared__ float attn[48 * 48];
    const int bh = blockIdx.x;
    const int b = bh >> 2, h = bh & 3;
    const float* ssq_q = sumsq + b * 384 + h * 48;
    const float* ssq_k = sumsq + b * 384 + CC + h * 48;
    const float temp = temperature[h];

    if (threadIdx.x < 48) {
        const int d = threadIdx.x;
        float rq = 1.0f / fmaxf(sqrtf(ssq_q[d]), 1e-12f);
        float row[48];
        float mx = -INFINITY;
        for (int e = 0; e < 48; ++e) {
            float s = 0.0f;
            for (int ks = 0; ks < KSPLIT; ++ks)
                s += Spart[((long long)ks * (BB * 4) + bh) * 2304 + d * 48 + e];
            float rk = 1.0f / fmaxf(sqrtf(ssq_k[e]), 1e-12f);
            float v = s * rq * rk * temp;
            row[e] = v;
            mx = fmaxf(mx, v);
        }
        float sum = 0.0f;
        for (int e = 0; e < 48; ++e) { float ex = __expf(row[e] - mx); row[e] = ex; sum += ex; }
        float inv = 1.0f / sum;
        for (int e = 0; e < 48; ++e) attn[d * 48 + e] = row[e] * inv;
    }
    __syncthreads();

    float* Mb = Mcomb + (long long)b * CC * CC;
    for (int idx = threadIdx.x; idx < CC * 48; idx += 256) {
        int o = idx / 48, e = idx - (idx / 48) * 48;
        float s = 0.0f;
        for (int d = 0; d < 48; ++d)
            s = fmaf(w_proj[o * CC + h * 48 + d], attn[d * 48 + e], s);
        Mb[o * CC + h * 48 + e] = s;   // disjoint column slices per h
    }
}

// ---------------------------------------------------------------------------
extern "C" void kernel_launch(void* const* d_in, const int* in_sizes, int n_in,
                              void* d_out, int out_size, void* d_ws, size_t ws_size,
                              hipStream_t stream)
{
    const float* x      = (const float*)d_in[0];   // (8,192,128,128)
    const float* w_qkv  = (const float*)d_in[1];   // (576,192)
    const float* w_dw   = (const float*)d_in[2];   // (576,1,3,3)
    const float* w_proj = (const float*)d_in[3];   // (192,192)
    const float* temp   = (const float*)d_in[4];   // (4,)
    float* out = (float*)d_out;                    // (8,192,128,128)

    float* ws = (float*)d_ws;
    const long long planeB = (long long)C3 * NN;   // per-batch qkv elements
    float* qkv1  = ws;                             // B*576*N
    float* qkvdw = qkv1 + (long long)BB * planeB;  // B*576*N
    float* sumsq = qkvdw + (long long)BB * planeB; // B*384
    float* Spart = sumsq + BB * 384;               // KSPLIT*B*4*48*48
    float* Mcomb = Spart + (long long)KSPLIT * BB * 4 * 48 * 48;  // B*192*192

    // K1: qkv1 = w_qkv (576x192) x x (192xN) per batch
    gemm_wmma_f32<<<dim3(NN / 256, C3 / 32, BB), 128, 0, stream>>>(
        w_qkv, 0LL, x, (long long)CC * NN, qkv1, (long long)C3 * NN,
        C3, NN, CC);

    // K2: depthwise 3x3, SAME
    dwconv3x3<<<dim3(NN / 256, C3, BB), 256, 0, stream>>>(qkv1, w_dw, qkvdw);

    // K3: per-row sum of squares for q and k channels
    rowsumsq<<<dim3(384, BB), 256, 0, stream>>>(qkvdw, sumsq);

    // K4: split-K Gram partials via WMMA (full 48x48 per wave)
    gram_wmma<<<dim3(KSPLIT, BB * 4), 32, 0, stream>>>(qkvdw, Spart);

    // K5: reduce + normalize + softmax + fold projection into Mcomb
    attn_fold<<<dim3(BB * 4), 256, 0, stream>>>(Spart, sumsq, temp, w_proj, Mcomb);

    // K6: out = Mcomb (192x192) x v (192xN) per batch  (v = qkvdw channels 384..575)
    gemm_wmma_f32<<<dim3(NN / 256, CC / 32, BB), 128, 0, stream>>>(
        Mcomb, (long long)CC * CC,
        qkvdw + (long long)2 * CC * NN, (long long)C3 * NN,
        out, (long long)CC * NN,
        CC, NN, CC);
}